// MMN_60911226192165
// MI455X (gfx1250) — compile-verified
//
#include <hip/hip_runtime.h>
#include <hip/hip_bf16.h>

// MMN dual-space proposal scoring for MI455X (gfx1250, wave32).
//
// Shapes: B=16 videos, C=256 channels, N=64 (proposal grid NxN=4096),
// 8 sentences/video, S=128 sentences total.
// out = [logits2d (S*N*N f32) | scores2d (S*N*N f32)]
//
// Roofline: 134 MB of read-once video features @ 23.3 TB/s ~ 6 us; compute
// (0.5 GFLOP) is noise. So: read each video element exactly once (per-video
// [4096 x 256] x [256 x 8] GEMM via V_WMMA_F32_16X16X4_F32, full f32),
// fold the per-proposal L2 norm in as a fused sum-of-squares post-scale,
// stream video/outputs with non-temporal hints, and coalesce the scattered
// per-sentence stores through a per-wave LDS transpose buffer.

typedef float v2f __attribute__((ext_vector_type(2)));
typedef float v8f __attribute__((ext_vector_type(8)));

#define NVID 16
#define NCH 256
#define NGRID 64
#define NN 4096          // NGRID*NGRID
#define SENTS 8
#define NSENT 128        // total sentences

__global__ __launch_bounds__(256)
void mmn_dualspace_kernel(const float* __restrict__ vf1,
                          const float* __restrict__ vf2,
                          const float* __restrict__ sf1,
                          const float* __restrict__ sf2,
                          const float* __restrict__ mask2d,
                          float* __restrict__ out_logits,
                          float* __restrict__ out_scores) {
    // sfn[space][n][c]: normalized sentence feats for this video, rows 8..15 = 0
    __shared__ float sfn[2][16][NCH];       // 32 KB
    // per-wave output-transpose buffer: [wave][logits|scores][sent][prop(+pad)]
    __shared__ float obuf[8][2][8][17];     // 8.5 KB, pad 17 kills bank conflicts

    const int tid  = threadIdx.x;
    const int wave = tid >> 5;          // 0..7
    const int lane = tid & 31;
    const int lo   = lane & 15;         // M / N index within fragment
    const int hi   = lane >> 4;         // half-wave select

    const int b  = blockIdx.x >> 4;     // video 0..15
    const int bi = blockIdx.x & 15;     // block-within-video 0..15

    // ---- Phase 1: normalize this video's 8 sentence vectors (both spaces) into LDS
    for (int sp = 0; sp < 2; ++sp) {
        const float* row = (sp == 0 ? sf1 : sf2) + (size_t)(b * SENTS + wave) * NCH;
        float v[8];
        float ss = 0.f;
#pragma unroll
        for (int k = 0; k < 8; ++k) {
            v[k] = row[lane + 32 * k];
            ss = fmaf(v[k], v[k], ss);
        }
#pragma unroll
        for (int off = 16; off >= 1; off >>= 1)
            ss += __shfl_xor(ss, off);
        const float inv = 1.0f / fmaxf(sqrtf(ss), 1e-12f);
#pragma unroll
        for (int k = 0; k < 8; ++k)
            sfn[sp][wave][lane + 32 * k] = v[k] * inv;
    }
    // zero pad rows 8..15 of both spaces (2*8*256 = 4096 floats, 16 per thread)
    for (int idx = tid; idx < 2 * 8 * NCH; idx += 256) {
        const int sp = idx >> 11;
        const int r8 = (idx >> 8) & 7;
        const int c  = idx & 255;
        sfn[sp][8 + r8][c] = 0.f;
    }
    __syncthreads();

    // ---- Phase 2: each wave processes 16-proposal tiles of this video.
    // 256 tiles/video over 16 blocks * 8 waves -> 2 tiles per wave.
    const size_t vbase = (size_t)b * NCH * NN;
    for (int rep = 0; rep < 2; ++rep) {
        const int tile = bi * 8 + wave + rep * 128;
        const int p0 = tile * 16;

        float raws[2][8];
#pragma unroll
        for (int sp = 0; sp < 2; ++sp) {
            // A fragment source: lane l reads vf[b, c0 + 2*hi + {0,1}, p0 + lo]
            const float* vptr = (sp == 0 ? vf1 : vf2) + vbase
                                + (size_t)(hi * 2) * NN + (p0 + lo);
            const float* bptr = &sfn[sp][lo][hi * 2];

            v8f acc = {0.f, 0.f, 0.f, 0.f, 0.f, 0.f, 0.f, 0.f};
            float na = 0.f;    // partial sum of squares (channels == 2*hi,2*hi+1 mod 4)
#pragma unroll 8
            for (int c0 = 0; c0 < NCH; c0 += 4) {
                v2f a;
                a.x = __builtin_nontemporal_load(vptr + c0 * NN);       // read-once stream
                a.y = __builtin_nontemporal_load(vptr + c0 * NN + NN);
                const v2f bb = *(const v2f*)(bptr + c0);                // ds_load_b64
                na = fmaf(a.x, a.x, na);
                na = fmaf(a.y, a.y, na);
                // D = A(16x4) * B(4x16) + C, full f32
                acc = __builtin_amdgcn_wmma_f32_16x16x4_f32(
                    false, a, false, bb, (short)0, acc, false, false);
            }
            // total ||v||^2 for proposal M=lo is split across half-waves
            const float nsum = na + __shfl_xor(na, 16);
            const float inv = 1.0f / fmaxf(sqrtf(nsum), 1e-12f);
            // D element r at lane l is (M = r + 8*hi, N = lo); fetch inv-norm of M
#pragma unroll
            for (int r = 0; r < 8; ++r)
                raws[sp][r] = acc[r] * __shfl(inv, r + 8 * hi);
        }

        // ---- fused elementwise epilogue -> per-wave LDS transpose
#pragma unroll
        for (int r = 0; r < 8; ++r) {
            const int m  = r + 8 * hi;                // proposal-in-tile, 0..15
            const int ij = p0 + m;                    // flat i*64+j
            const float mk = mask2d[ij];
            const float logit = raws[0][r] * 10.f;
            const float iou = mk / (1.f + __expf(-logit));
            const float con = (raws[1][r] + 1.f) * 0.5f * mk;
            const float sc  = sqrtf(con) * iou;
            if (lo < 8) {                             // N index = sentence-in-video
                obuf[wave][0][lo][m] = logit;
                obuf[wave][1][lo][m] = sc;
            }
        }
        // ---- coalesced stores: lane covers (s = 2k+hi, p = lo); each store is
        // two contiguous 64B half-wave segments. Same-wave DS ordering, no barrier.
#pragma unroll
        for (int k = 0; k < 4; ++k) {
            const int s = 2 * k + hi;
            const float lg  = obuf[wave][0][s][lo];
            const float scv = obuf[wave][1][s][lo];
            const size_t o = (size_t)((b << 3) + s) * NN + (p0 + lo);
            __builtin_nontemporal_store(lg,  &out_logits[o]);
            __builtin_nontemporal_store(scv, &out_scores[o]);
        }
    }
}

extern "C" void kernel_launch(void* const* d_in, const int* in_sizes, int n_in,
                              void* d_out, int out_size, void* d_ws, size_t ws_size,
                              hipStream_t stream) {
    const float* vf1    = (const float*)d_in[0];  // [16,256,64,64]
    const float* vf2    = (const float*)d_in[1];  // [16,256,64,64]
    const float* sf1    = (const float*)d_in[2];  // [128,256]
    const float* sf2    = (const float*)d_in[3];  // [128,256]
    const float* mask2d = (const float*)d_in[4];  // [64,64]
    // d_in[5] = num_sentences (uniform 8 per the setup; layout baked in)

    float* out_logits = (float*)d_out;
    float* out_scores = out_logits + (size_t)NSENT * NN;

    dim3 grid(NVID * 16);   // 16 videos x 16 blocks
    dim3 block(256);        // 8 waves of 32
    mmn_dualspace_kernel<<<grid, block, 0, stream>>>(
        vf1, vf2, sf1, sf2, mask2d, out_logits, out_scores);
}